// GCN_8280696947369
// MI455X (gfx1250) — compile-verified
//
#include <hip/hip_runtime.h>

#define NN 50000
#define EE 800000
#define FIN 128
#define HH 96
#define CC 10
#define GG 128

typedef float v2f __attribute__((ext_vector_type(2)));
typedef float v8f __attribute__((ext_vector_type(8)));

// ---------------- utility ----------------
__global__ void fill_kernel(float* __restrict__ p, int n, float v) {
    int t = blockIdx.x * blockDim.x + threadIdx.x;
    if (t < n) p[t] = v;
}

// deg[dst_f[e]] += 1 over E real edges + N self loops
__global__ void degree_kernel(const int* __restrict__ dst, float* __restrict__ deg, int total) {
    int t = blockIdx.x * blockDim.x + threadIdx.x;
    if (t >= total) return;
    int d = (t < EE) ? dst[t] : (t - EE);
    atomicAdd(&deg[d], 1.0f);
}

// dis = deg > 0 ? rsqrt(deg) : 0   (in place)
__global__ void rsqrt_kernel(float* __restrict__ deg) {
    int t = blockIdx.x * blockDim.x + threadIdx.x;
    if (t < NN) {
        float d = deg[t];
        deg[t] = (d > 0.0f) ? rsqrtf(d) : 0.0f;
    }
}

// ---------------- fp32 WMMA GEMM: C[M x 96] = A[M x K] * B[K x 96] ----------------
// One wave32 computes one 16x16 output tile via V_WMMA_F32_16X16X4_F32.
// A frag (16x4):  lane<16 -> row=lane,    VGPR{0,1}=K{0,1}; lane>=16 -> K{2,3}
// B frag (4x16):  lane<16 -> col=lane,    VGPR{0,1}=K{0,1}; lane>=16 -> K{2,3}
// C frag:         VGPR v -> row = v + 8*(lane>=16), col = lane&15
__global__ void gemm_wmma_kernel(const float* __restrict__ A, const float* __restrict__ B,
                                 float* __restrict__ C, int M, int K) {
    const int tilesN = HH / 16; // 6
    int wid = blockIdx.x * (blockDim.x >> 5) + (threadIdx.x >> 5);
    int tileM = wid / tilesN;
    int tileN = wid - tileM * tilesN;
    if (tileM * 16 >= M) return; // wave-uniform: EXEC stays all-ones for WMMA

    int lane = threadIdx.x & 31;
    int hv   = lane >> 4;   // 0: K pair {0,1}, 1: K pair {2,3}
    int r    = lane & 15;

    const float* arow = A + (size_t)(tileM * 16 + r) * K + 2 * hv;
    const float* bp   = B + (size_t)(2 * hv) * HH + tileN * 16 + r;

    v8f acc = {};
    for (int k = 0; k < K; k += 4) {
        v2f a, b;
        a.x = arow[0];  a.y = arow[1];
        b.x = bp[0];    b.y = bp[HH];
        acc = __builtin_amdgcn_wmma_f32_16x16x4_f32(
            /*neg_a=*/false, a, /*neg_b=*/false, b,
            /*c_mod=*/(short)0, acc, /*reuse_a=*/false, /*reuse_b=*/false);
        arow += 4;
        bp   += 4 * HH;
    }

    float* crow = C + (size_t)(tileM * 16 + hv * 8) * HH + tileN * 16 + r;
#pragma unroll
    for (int v = 0; v < 8; ++v) crow[(size_t)v * HH] = acc[v];
}

// ---------------- aggregation ----------------
// agg[i] = tmp[i] * dis[i]^2   (self-loop term, atomic-free initializer)
__global__ void selfloop_kernel(const float* __restrict__ t, const float* __restrict__ dis,
                                float* __restrict__ agg) {
    int tid = blockIdx.x * blockDim.x + threadIdx.x;
    if (tid >= NN * HH) return;
    int i = tid / HH;
    float w = dis[i];
    agg[tid] = t[tid] * (w * w);
}

// agg[dst] += tmp[src] * dis[src]*dis[dst]; one thread per (edge, float4 chunk)
__global__ void edge_scatter_kernel(const int* __restrict__ src, const int* __restrict__ dst,
                                    const float* __restrict__ dis, const float* __restrict__ t,
                                    float* __restrict__ agg) {
    int tid = blockIdx.x * blockDim.x + threadIdx.x;
    const int chunks = HH / 4; // 24
    if (tid >= EE * chunks) return;
    int e = tid / chunks;
    int c = tid - e * chunks;
    int s = src[e];
    int d = dst[e];
    float w = dis[s] * dis[d];
    const float4 v = *(const float4*)(t + (size_t)s * HH + c * 4);
    float* p = agg + (size_t)d * HH + c * 4;
    atomicAdd(p + 0, v.x * w);
    atomicAdd(p + 1, v.y * w);
    atomicAdd(p + 2, v.z * w);
    atomicAdd(p + 3, v.w * w);
}

// h = relu(h + bias[f])
__global__ void bias_relu_kernel(float* __restrict__ h, const float* __restrict__ bias) {
    int tid = blockIdx.x * blockDim.x + threadIdx.x;
    if (tid >= NN * HH) return;
    int f = tid % HH;
    h[tid] = fmaxf(h[tid] + bias[f], 0.0f);
}

// ---------------- pooling + MLP head ----------------
__global__ void count_kernel(const int* __restrict__ batch, float* __restrict__ cnt) {
    int t = blockIdx.x * blockDim.x + threadIdx.x;
    if (t < NN) atomicAdd(&cnt[batch[t]], 1.0f);
}

__global__ void pool_kernel(const int* __restrict__ batch, const float* __restrict__ h,
                            float* __restrict__ pooled) {
    int tid = blockIdx.x * blockDim.x + threadIdx.x;
    if (tid >= NN * HH) return;
    int i = tid / HH;
    int f = tid - i * HH;
    atomicAdd(&pooled[(size_t)batch[i] * HH + f], h[tid]);
}

// hid[g][j] = relu(bl1[j] + (sum_k pooled[g][k]*Wl1[k][j]) / max(cnt[g],1))
__global__ void head1_kernel(const float* __restrict__ pooled, const float* __restrict__ cnt,
                             const float* __restrict__ Wl1, const float* __restrict__ bl1,
                             float* __restrict__ hid) {
    int tid = blockIdx.x * blockDim.x + threadIdx.x;
    if (tid >= GG * HH) return;
    int g = tid / HH;
    int j = tid - g * HH;
    float s = 0.0f;
    const float* pr = pooled + (size_t)g * HH;
#pragma unroll 8
    for (int k = 0; k < HH; ++k) s = fmaf(pr[k], Wl1[(size_t)k * HH + j], s);
    float inv = 1.0f / fmaxf(cnt[g], 1.0f);
    hid[tid] = fmaxf(fmaf(s, inv, bl1[j]), 0.0f);
}

// out[g][c] = bl2[c] + sum_j hid[g][j]*Wl2[j][c]
__global__ void head2_kernel(const float* __restrict__ hid, const float* __restrict__ Wl2,
                             const float* __restrict__ bl2, float* __restrict__ out) {
    int tid = blockIdx.x * blockDim.x + threadIdx.x;
    if (tid >= GG * CC) return;
    int g = tid / CC;
    int c = tid - g * CC;
    float s = bl2[c];
    const float* hr = hid + (size_t)g * HH;
#pragma unroll 8
    for (int j = 0; j < HH; ++j) s = fmaf(hr[j], Wl2[(size_t)j * CC + c], s);
    out[tid] = s;
}

extern "C" void kernel_launch(void* const* d_in, const int* in_sizes, int n_in,
                              void* d_out, int out_size, void* d_ws, size_t ws_size,
                              hipStream_t stream) {
    const float* x     = (const float*)d_in[0];
    const int*   ei    = (const int*)d_in[1];
    const int*   batch = (const int*)d_in[2];
    const float* W1    = (const float*)d_in[3];
    const float* b1    = (const float*)d_in[4];
    const float* Wh    = (const float*)d_in[5];
    const float* bh    = (const float*)d_in[6];
    const float* Wl1   = (const float*)d_in[7];
    const float* bl1   = (const float*)d_in[8];
    const float* Wl2   = (const float*)d_in[9];
    const float* bl2   = (const float*)d_in[10];
    float* out = (float*)d_out;

    const int* src = ei;
    const int* dst = ei + EE;

    // workspace layout (floats)
    float* ws     = (float*)d_ws;
    float* dis    = ws;                      // N     (deg -> rsqrt in place)
    float* tmp    = dis + NN;                // N*H
    float* hcur   = tmp + (size_t)NN * HH;   // N*H
    float* cnt    = hcur + (size_t)NN * HH;  // G
    float* pooled = cnt + GG;                // G*H
    float* hid    = pooled + (size_t)GG * HH;// G*H

    const int T = 256;

    // symmetric normalization coefficients
    fill_kernel<<<(NN + T - 1) / T, T, 0, stream>>>(dis, NN, 0.0f);
    degree_kernel<<<((EE + NN) + T - 1) / T, T, 0, stream>>>(dst, dis, EE + NN);
    rsqrt_kernel<<<(NN + T - 1) / T, T, 0, stream>>>(dis);

    const int tiles  = (NN / 16) * (HH / 16);      // 3125 * 6 = 18750 waves
    const int gblk   = (tiles + 7) / 8;            // 8 waves / 256-thread block
    const int nhGrid = (NN * HH + T - 1) / T;
    const int esGrid = (EE * (HH / 4) + T - 1) / T;

    for (int l = 0; l < 4; ++l) {
        const float* A = (l == 0) ? x : hcur;
        int K          = (l == 0) ? FIN : HH;
        const float* W = (l == 0) ? W1 : Wh + (size_t)(l - 1) * HH * HH;
        const float* b = (l == 0) ? b1 : bh + (size_t)(l - 1) * HH;

        gemm_wmma_kernel<<<gblk, T, 0, stream>>>(A, W, tmp, NN, K);
        selfloop_kernel<<<nhGrid, T, 0, stream>>>(tmp, dis, hcur);
        edge_scatter_kernel<<<esGrid, T, 0, stream>>>(src, dst, dis, tmp, hcur);
        bias_relu_kernel<<<nhGrid, T, 0, stream>>>(hcur, b);
    }

    // global mean pool + MLP head
    fill_kernel<<<1, T, 0, stream>>>(cnt, GG, 0.0f);
    fill_kernel<<<(GG * HH + T - 1) / T, T, 0, stream>>>(pooled, GG * HH, 0.0f);
    count_kernel<<<(NN + T - 1) / T, T, 0, stream>>>(batch, cnt);
    pool_kernel<<<nhGrid, T, 0, stream>>>(batch, hcur, pooled);
    head1_kernel<<<(GG * HH + T - 1) / T, T, 0, stream>>>(pooled, cnt, Wl1, bl1, hid);
    head2_kernel<<<(GG * CC + T - 1) / T, T, 0, stream>>>(hid, Wl2, bl2, out);
}